// Attention_15109694948045
// MI455X (gfx1250) — compile-verified
//
#include <hip/hip_runtime.h>
#include <math.h>

#define NB 16          // batch
#define SECL 16
#define WORDL 256
#define SRCL (SECL*WORDL)   // 4096
#define DIM 1024
#define FSZ 2
#define NROWS (NB*FSZ*WORDL)        // 8192 selected rows total
#define GATH (FSZ*WORDL)            // 512 gathered entries per batch
#define NCHUNK 8
#define RPC (GATH/NCHUNK)           // 64 rows per chunk

typedef float v2f __attribute__((ext_vector_type(2)));
typedef float v8f __attribute__((ext_vector_type(8)));

// ---------------------------------------------------------------- top-k(focus,2)
__global__ void __launch_bounds__(32)
topk_kernel(const float* __restrict__ focus, int* __restrict__ top_idx,
            float* __restrict__ top_val) {
    int b = threadIdx.x;
    if (b >= NB) return;
    const float* f = focus + b * SECL;
    int i0 = 0; float v0 = f[0];
    for (int i = 1; i < SECL; ++i) { float x = f[i]; if (x > v0) { v0 = x; i0 = i; } }
    int i1 = (i0 == 0) ? 1 : 0; float v1 = f[i1];
    for (int i = 0; i < SECL; ++i) {
        if (i == i0) continue;
        float x = f[i];
        if (x > v1) { v1 = x; i1 = i; }
    }
    top_idx[b*2+0] = i0; top_idx[b*2+1] = i1;
    top_val[b*2+0] = v0; top_val[b*2+1] = v1;
}

// ------------------------------------------- dec_feature = dec_hidden @ W^T + b
// One wave per 16-column output tile; V_WMMA_F32_16X16X4_F32 over K=1024.
__global__ void __launch_bounds__(32)
decproj_wmma_kernel(const float* __restrict__ A,      // dec_hidden [16][1024]
                    const float* __restrict__ W,      // W_dec [1024][1024]
                    const float* __restrict__ bias,   // [1024]
                    float* __restrict__ dec_feature)  // [16][1024]
{
    const int n0   = blockIdx.x * 16;
    const int lane = threadIdx.x;       // 0..31
    const int half = lane >> 4;         // 0/1 -> K sub-pair
    const int lx   = lane & 15;         // M (for A) / N (for B)

    v8f c = {0.f,0.f,0.f,0.f,0.f,0.f,0.f,0.f};
    const float* arow = A + lx * DIM + 2 * half;          // A[lx][k+2*half ..]
    const float* brow = W + (size_t)(n0 + lx) * DIM + 2 * half; // W[n][k] = B[k][n]

    for (int k = 0; k < DIM; k += 4) {
        v2f a = *(const v2f*)(arow + k);
        v2f bm = *(const v2f*)(brow + k);
        c = __builtin_amdgcn_wmma_f32_16x16x4_f32(false, a, false, bm,
                                                  (short)0, c, false, false);
    }
    float bb = bias[n0 + lx];
    #pragma unroll
    for (int r = 0; r < 8; ++r) {
        int m = r + 8 * half;                    // C/D layout: VGPR r, half-wave
        dec_feature[m * DIM + n0 + lx] = c[r] + bb;
    }
}

// ----------------------- scores for selected sections only (wave per src row)
__global__ void __launch_bounds__(256)
score_kernel(const float* __restrict__ enc_feature,   // [B][S][D]
             const float* __restrict__ coverage,      // [B][S]
             const float* __restrict__ dec_feature,   // [B][D]
             const float* __restrict__ v,             // [D]
             const float* __restrict__ w_cov,         // [D]
             const int*   __restrict__ top_idx,       // [B][2]
             float* __restrict__ score_g)             // [B][512]
{
    int wave = (blockIdx.x * blockDim.x + threadIdx.x) >> 5;
    int lane = threadIdx.x & 31;
    if (wave >= NROWS) return;
    int b    = wave >> 9;        // / 512
    int rem  = wave & 511;
    int ksec = rem >> 8;
    int w    = rem & 255;
    int s    = top_idx[b*2 + ksec] * WORDL + w;

    const float4* ef = (const float4*)(enc_feature + ((size_t)b * SRCL + s) * DIM);
    const float4* df = (const float4*)(dec_feature + b * DIM);
    const float4* vv = (const float4*)v;
    const float4* wc = (const float4*)w_cov;
    float cov = coverage[b * SRCL + s];

    float acc = 0.f;
    #pragma unroll
    for (int i = 0; i < 8; ++i) {
        int idx = i * 32 + lane;                 // coalesced float4 across wave
        float4 e = ef[idx], d = df[idx], vd = vv[idx], c = wc[idx];
        acc += vd.x * tanhf(e.x + d.x + cov * c.x);
        acc += vd.y * tanhf(e.y + d.y + cov * c.y);
        acc += vd.z * tanhf(e.z + d.z + cov * c.z);
        acc += vd.w * tanhf(e.w + d.w + cov * c.w);
    }
    #pragma unroll
    for (int off = 16; off > 0; off >>= 1) acc += __shfl_xor(acc, off, 32);
    if (lane == 0) score_g[b * GATH + rem] = acc;
}

// ---- softmax chain collapses to attn_i = f_sec * e^{s_i-max} * m_i / total --
__global__ void __launch_bounds__(256)
softmax_kernel(const float* __restrict__ score_g,
               const float* __restrict__ enc_mask,   // [B][S]
               const int*   __restrict__ top_idx,
               const float* __restrict__ top_val,
               float* __restrict__ attn_g)           // [B][512]
{
    __shared__ float red[256];
    int b = blockIdx.x, t = threadIdx.x;
    float sc0 = score_g[b * GATH + t];
    float sc1 = score_g[b * GATH + 256 + t];

    red[t] = fmaxf(sc0, sc1); __syncthreads();
    for (int o = 128; o > 0; o >>= 1) { if (t < o) red[t] = fmaxf(red[t], red[t+o]); __syncthreads(); }
    float mx = red[0]; __syncthreads();

    int   sec0 = top_idx[b*2+0], sec1 = top_idx[b*2+1];
    float f0 = top_val[b*2+0],  f1 = top_val[b*2+1];
    float m0 = enc_mask[b * SRCL + sec0 * WORDL + t];
    float m1 = enc_mask[b * SRCL + sec1 * WORDL + t];
    float e0 = __expf(sc0 - mx) * m0 * f0;
    float e1 = __expf(sc1 - mx) * m1 * f1;

    red[t] = e0 + e1; __syncthreads();
    for (int o = 128; o > 0; o >>= 1) { if (t < o) red[t] += red[t+o]; __syncthreads(); }
    float inv = 1.0f / red[0];
    attn_g[b * GATH + t]       = e0 * inv;
    attn_g[b * GATH + 256 + t] = e1 * inv;
}

// ------------------- scatter attn_dist + coverage_out over full [B,S] --------
__global__ void __launch_bounds__(256)
scatter_kernel(const float* __restrict__ coverage,
               const int*   __restrict__ top_idx,
               const float* __restrict__ attn_g,
               float* __restrict__ out_attn,
               float* __restrict__ out_cov)
{
    int gid = blockIdx.x * 256 + threadIdx.x;  // 0..65535
    int b = gid >> 12, s = gid & 4095;
    int sec = s >> 8, w = s & 255;
    float val = 0.f;
    if      (sec == top_idx[b*2+0]) val = attn_g[b * GATH + w];
    else if (sec == top_idx[b*2+1]) val = attn_g[b * GATH + 256 + w];
    out_attn[gid] = val;
    out_cov[gid]  = coverage[gid] + val;
}

// ------------ context partials: deterministic split over 8 row-chunks --------
__global__ void __launch_bounds__(256)
context_partial_kernel(const float* __restrict__ enc_output,  // [B][S][D]
                       const int*   __restrict__ top_idx,
                       const float* __restrict__ attn_g,
                       float* __restrict__ partial)           // [B][8][D]
{
    __shared__ float a_sh[RPC];
    __shared__ int   s_sh[RPC];
    int blk   = blockIdx.x;          // 16 * 8 * 4 = 512 blocks
    int dq    = blk & 3;
    int chunk = (blk >> 2) & 7;
    int b     = blk >> 5;
    int t     = threadIdx.x;
    if (t < RPC) {
        int row  = chunk * RPC + t;            // 0..511 gathered index
        int ksec = row >> 8, w = row & 255;
        a_sh[t] = attn_g[b * GATH + row];
        s_sh[t] = top_idx[b*2 + ksec] * WORDL + w;
    }
    __syncthreads();
    int d = dq * 256 + t;
    const float* base = enc_output + (size_t)b * SRCL * DIM + d;
    float acc = 0.f;
    for (int r = 0; r < RPC; ++r) {
        if (r + 4 < RPC)
            __builtin_prefetch(base + (size_t)s_sh[r + 4] * DIM, 0, 0); // global_prefetch_b8
        acc += a_sh[r] * base[(size_t)s_sh[r] * DIM];
    }
    partial[((size_t)b * NCHUNK + chunk) * DIM + d] = acc;
}

__global__ void __launch_bounds__(256)
context_reduce_kernel(const float* __restrict__ partial, float* __restrict__ out_ctx)
{
    int gid = blockIdx.x * 256 + threadIdx.x;   // 0..16383
    int b = gid >> 10, d = gid & 1023;
    float acc = 0.f;
    #pragma unroll
    for (int c = 0; c < NCHUNK; ++c) acc += partial[(b * NCHUNK + c) * DIM + d];
    out_ctx[gid] = acc;
}

// ---------------------------------------------------------------------------
extern "C" void kernel_launch(void* const* d_in, const int* in_sizes, int n_in,
                              void* d_out, int out_size, void* d_ws, size_t ws_size,
                              hipStream_t stream) {
    const float* dec_hidden  = (const float*)d_in[0];
    const float* enc_output  = (const float*)d_in[1];
    const float* enc_feature = (const float*)d_in[2];
    const float* enc_mask    = (const float*)d_in[3];
    // d_in[4] = sec_attn : unused by the reference computation
    const float* coverage    = (const float*)d_in[5];
    const float* focus       = (const float*)d_in[6];
    const float* W_dec       = (const float*)d_in[7];
    const float* b_dec       = (const float*)d_in[8];
    const float* v           = (const float*)d_in[9];
    const float* w_cov       = (const float*)d_in[10];

    float* ws = (float*)d_ws;
    float* ws_decf  = ws;                       // 16384
    int*   ws_topi  = (int*)(ws + 16384);       // 32 ints
    float* ws_topf  = ws + 16384 + 32;          // 32
    float* ws_score = ws + 16448;               // 8192
    float* ws_attn  = ws + 24640;               // 8192
    float* ws_part  = ws + 32832;               // 131072  (end: ~656 KB)

    float* out_ctx  = (float*)d_out;            // [16,1024]
    float* out_attn = out_ctx + NB * DIM;       // [16,4096]
    float* out_cov  = out_attn + NB * SRCL;     // [16,4096]

    topk_kernel<<<1, 32, 0, stream>>>(focus, ws_topi, ws_topf);
    decproj_wmma_kernel<<<DIM / 16, 32, 0, stream>>>(dec_hidden, W_dec, b_dec, ws_decf);
    score_kernel<<<NROWS / 8, 256, 0, stream>>>(enc_feature, coverage, ws_decf,
                                                v, w_cov, ws_topi, ws_score);
    softmax_kernel<<<NB, 256, 0, stream>>>(ws_score, enc_mask, ws_topi, ws_topf, ws_attn);
    scatter_kernel<<<(NB * SRCL) / 256, 256, 0, stream>>>(coverage, ws_topi, ws_attn,
                                                          out_attn, out_cov);
    context_partial_kernel<<<NB * NCHUNK * 4, 256, 0, stream>>>(enc_output, ws_topi,
                                                                ws_attn, ws_part);
    context_reduce_kernel<<<(NB * DIM) / 256, 256, 0, stream>>>(ws_part, out_ctx);
}